// LineWrapper_25786983645443
// MI455X (gfx1250) — compile-verified
//
#include <hip/hip_runtime.h>

typedef __attribute__((ext_vector_type(2))) float v2f;
typedef __attribute__((ext_vector_type(8))) float v8f;

#define F 32
#define TB 256

__device__ __forceinline__ float relu_f(float v) { return v > 0.0f ? v : 0.0f; }

// ---------------------------------------------------------------------------
// Kernel 1: new_x = 0.5*(edge_attr[e] + edge_attr[e+E]); zero accumulators.
// ---------------------------------------------------------------------------
__global__ __launch_bounds__(TB) void k_init(
    const float* __restrict__ edge_attr, float* __restrict__ new_x,
    float* __restrict__ agg, float* __restrict__ degE,
    float* __restrict__ xacc, float* __restrict__ ncnt, int E, int N) {
  int i = blockIdx.x * TB + threadIdx.x;
  int EF = E * F, NF = N * F;
  if (i < EF) {
    new_x[i] = 0.5f * (edge_attr[i] + edge_attr[i + EF]);
    agg[i] = 0.0f;
  }
  if (i < NF) xacc[i] = 0.0f;
  if (i < E) degE[i] = 0.0f;
  if (i < N) ncnt[i] = 0.0f;
}

// ---------------------------------------------------------------------------
// Kernel 2: degE[lg_dst[i]] += 1 ; ncnt[lg_node[i]] += 1   (i in [0,L))
// ---------------------------------------------------------------------------
__global__ __launch_bounds__(TB) void k_counts(
    const int* __restrict__ lg_dst, const int* __restrict__ lg_node,
    float* __restrict__ degE, float* __restrict__ ncnt, int L) {
  int i = blockIdx.x * TB + threadIdx.x;
  if (i < L) {
    atomicAdd(&degE[lg_dst[i]], 1.0f);
    atomicAdd(&ncnt[lg_node[i]], 1.0f);
  }
}

// ---------------------------------------------------------------------------
// Kernel 3 (main, WMMA): per 16-row tile of the line-graph incidence list:
//   rowA = new_x[lg_src] + x[lg_node] ; msg = relu(rowA @ Wmsg) -> atomicAdd agg[lg_dst]
//   rowX = x[lg_node]                 ; eo  = relu(rowX @ We)   -> atomicAdd xacc[lg_node]
// A fragment (16x4 f32): lanes 0-15 row M=lm {K0 in v.x, K1 in v.y}; lanes 16-31 {K2,K3}.
// B fragment (4x16 f32): v.x = W[K base + 2*half][col=lm(+16n)], v.y = next K row.
// C/D (16x16 f32): vgpr j -> row M=j+8*half, col N=lm.
// ---------------------------------------------------------------------------
__global__ __launch_bounds__(TB) void k_msg(
    const float* __restrict__ x, const float* __restrict__ new_x,
    const float* __restrict__ Wmsg, const float* __restrict__ We,
    const int* __restrict__ lg_src, const int* __restrict__ lg_dst,
    const int* __restrict__ lg_node,
    float* __restrict__ agg, float* __restrict__ xacc, int L) {
  __shared__ float sWm[F * F];
  __shared__ float sWe[F * F];
  for (int i = threadIdx.x; i < F * F; i += TB) {
    sWm[i] = Wmsg[i];
    sWe[i] = We[i];
  }
  __syncthreads();

  const int lane = threadIdx.x & 31;
  const int half = lane >> 4;   // 0: K pair {0,1}, rows M 0..7 ; 1: {2,3}, rows 8..15
  const int lm = lane & 15;

  const long wave = (long)blockIdx.x * (TB / 32) + (threadIdx.x >> 5);
  const long nWaves = (long)gridDim.x * (TB / 32);
  const long nTiles = ((long)L + 15) >> 4;

  for (long t = wave; t < nTiles; t += nWaves) {
    const int i0 = (int)(t << 4);
    // branchless clamp for the tail tile (EXEC must stay all-ones for WMMA)
    int myRow = i0 + lm;
    int cl = myRow < L ? myRow : (L - 1);
    const float* nxp = new_x + (long)lg_src[cl] * F;
    const float* xp = x + (long)lg_node[cl] * F;

    v2f am[8], ax[8];
#pragma unroll
    for (int kk = 0; kk < 8; ++kk) {
      const int c0 = 4 * kk + 2 * half;
      v2f xv;
      xv.x = xp[c0];
      xv.y = xp[c0 + 1];
      v2f nv;
      nv.x = nxp[c0];
      nv.y = nxp[c0 + 1];
      ax[kk] = xv;
      am[kk] = nv + xv;
    }

    v8f cm0 = {}, cm1 = {}, ce0 = {}, ce1 = {};
#pragma unroll
    for (int kk = 0; kk < 8; ++kk) {
      const int r0 = (4 * kk + 2 * half) * F;
      v2f bm0, bm1, be0, be1;
      bm0.x = sWm[r0 + lm];
      bm0.y = sWm[r0 + F + lm];
      bm1.x = sWm[r0 + 16 + lm];
      bm1.y = sWm[r0 + F + 16 + lm];
      be0.x = sWe[r0 + lm];
      be0.y = sWe[r0 + F + lm];
      be1.x = sWe[r0 + 16 + lm];
      be1.y = sWe[r0 + F + 16 + lm];
      cm0 = __builtin_amdgcn_wmma_f32_16x16x4_f32(false, am[kk], false, bm0,
                                                  (short)0, cm0, false, false);
      cm1 = __builtin_amdgcn_wmma_f32_16x16x4_f32(false, am[kk], false, bm1,
                                                  (short)0, cm1, false, false);
      ce0 = __builtin_amdgcn_wmma_f32_16x16x4_f32(false, ax[kk], false, be0,
                                                  (short)0, ce0, false, false);
      ce1 = __builtin_amdgcn_wmma_f32_16x16x4_f32(false, ax[kk], false, be1,
                                                  (short)0, ce1, false, false);
    }

#pragma unroll
    for (int j = 0; j < 8; ++j) {
      const int r = i0 + j + 8 * half;
      if (r < L) {
        const long d = (long)lg_dst[r] * F;
        const long nd = (long)lg_node[r] * F;
        atomicAdd(&agg[d + lm], relu_f(cm0[j]));
        atomicAdd(&agg[d + 16 + lm], relu_f(cm1[j]));
        atomicAdd(&xacc[nd + lm], relu_f(ce0[j]));
        atomicAdd(&xacc[nd + 16 + lm], relu_f(ce1[j]));
      }
    }
  }
}

// ---------------------------------------------------------------------------
// Kernel 4 (WMMA): x_line = relu(new_x @ Wn + agg / max(degE,1))
// ---------------------------------------------------------------------------
__global__ __launch_bounds__(TB) void k_xline(
    const float* __restrict__ new_x, const float* __restrict__ agg,
    const float* __restrict__ degE, const float* __restrict__ Wn,
    float* __restrict__ x_line, int E) {
  __shared__ float sW[F * F];
  for (int i = threadIdx.x; i < F * F; i += TB) sW[i] = Wn[i];
  __syncthreads();

  const int lane = threadIdx.x & 31;
  const int half = lane >> 4;
  const int lm = lane & 15;

  const long wave = (long)blockIdx.x * (TB / 32) + (threadIdx.x >> 5);
  const long nWaves = (long)gridDim.x * (TB / 32);
  const long nTiles = ((long)E + 15) >> 4;

  for (long t = wave; t < nTiles; t += nWaves) {
    const int i0 = (int)(t << 4);
    int myRow = i0 + lm;
    int cl = myRow < E ? myRow : (E - 1);
    const float* np = new_x + (long)cl * F;

    v2f a[8];
#pragma unroll
    for (int kk = 0; kk < 8; ++kk) {
      const int c0 = 4 * kk + 2 * half;
      v2f v;
      v.x = np[c0];
      v.y = np[c0 + 1];
      a[kk] = v;
    }

    v8f c0v = {}, c1v = {};
#pragma unroll
    for (int kk = 0; kk < 8; ++kk) {
      const int r0 = (4 * kk + 2 * half) * F;
      v2f b0, b1;
      b0.x = sW[r0 + lm];
      b0.y = sW[r0 + F + lm];
      b1.x = sW[r0 + 16 + lm];
      b1.y = sW[r0 + F + 16 + lm];
      c0v = __builtin_amdgcn_wmma_f32_16x16x4_f32(false, a[kk], false, b0,
                                                  (short)0, c0v, false, false);
      c1v = __builtin_amdgcn_wmma_f32_16x16x4_f32(false, a[kk], false, b1,
                                                  (short)0, c1v, false, false);
    }

#pragma unroll
    for (int j = 0; j < 8; ++j) {
      const int r = i0 + j + 8 * half;
      if (r < E) {
        float dv = degE[r];
        float inv = 1.0f / (dv > 1.0f ? dv : 1.0f);
        const long rb = (long)r * F;
        x_line[rb + lm] = relu_f(c0v[j] + agg[rb + lm] * inv);
        x_line[rb + 16 + lm] = relu_f(c1v[j] + agg[rb + 16 + lm] * inv);
      }
    }
  }
}

// ---------------------------------------------------------------------------
// Kernel 5: x_out = xacc / max(ncnt,1)   ;   edge_attr_out = x_line[rev_pair]
// d_out layout: [0, N*F) x_out, then [N*F, N*F + 2*E*F) edge_attr_out.
// ---------------------------------------------------------------------------
__global__ __launch_bounds__(TB) void k_final(
    const float* __restrict__ x_line, const float* __restrict__ xacc,
    const float* __restrict__ ncnt, const int* __restrict__ rev_pair,
    float* __restrict__ out, int N, int E) {
  int i = blockIdx.x * TB + threadIdx.x;
  const int NF = N * F;
  const int EO = 2 * E * F;
  if (i < NF) {
    float c = ncnt[i / F];
    out[i] = xacc[i] / (c > 1.0f ? c : 1.0f);
  }
  if (i < EO) {
    const int j = i / F;
    const int col = i - j * F;
    out[NF + i] = x_line[(long)rev_pair[j] * F + col];
  }
}

extern "C" void kernel_launch(void* const* d_in, const int* in_sizes, int n_in,
                              void* d_out, int out_size, void* d_ws,
                              size_t ws_size, hipStream_t stream) {
  const float* x = (const float*)d_in[0];
  const float* edge_attr = (const float*)d_in[1];
  const float* Wmsg = (const float*)d_in[2];
  const float* Wn = (const float*)d_in[3];
  const float* We = (const float*)d_in[4];
  // d_in[5] = pair_id (unused: each pair id appears exactly twice -> cnt == 2)
  const int* lg_src = (const int*)d_in[6];
  const int* lg_dst = (const int*)d_in[7];
  const int* lg_node = (const int*)d_in[8];
  const int* rev_pair = (const int*)d_in[9];

  const int N = in_sizes[0] / F;
  const int E = in_sizes[1] / (2 * F);
  const int L = in_sizes[6];
  const int EF = E * F, NF = N * F;

  // workspace layout (floats): new_x[EF] | agg[EF] | x_line[EF] | xacc[NF] | degE[E] | ncnt[N]
  float* ws = (float*)d_ws;
  float* new_x = ws;
  float* agg = new_x + EF;
  float* x_line = agg + EF;
  float* xacc = x_line + EF;
  float* degE = xacc + NF;
  float* ncnt = degE + E;

  {
    int n = EF > NF ? EF : NF;
    k_init<<<(n + TB - 1) / TB, TB, 0, stream>>>(edge_attr, new_x, agg, degE,
                                                 xacc, ncnt, E, N);
  }
  k_counts<<<(L + TB - 1) / TB, TB, 0, stream>>>(lg_dst, lg_node, degE, ncnt, L);
  {
    long nTiles = ((long)L + 15) / 16;
    int blocks = (int)((nTiles + (TB / 32) - 1) / (TB / 32));
    k_msg<<<blocks, TB, 0, stream>>>(x, new_x, Wmsg, We, lg_src, lg_dst,
                                     lg_node, agg, xacc, L);
  }
  {
    long nTiles = ((long)E + 15) / 16;
    int blocks = (int)((nTiles + (TB / 32) - 1) / (TB / 32));
    k_xline<<<blocks, TB, 0, stream>>>(new_x, agg, degE, Wn, x_line, E);
  }
  {
    int n = (2 * EF > NF) ? 2 * EF : NF;
    k_final<<<(n + TB - 1) / TB, TB, 0, stream>>>(x_line, xacc, ncnt, rev_pair,
                                                  (float*)d_out, N, E);
  }
}